// WienerFilter_21827023798919
// MI455X (gfx1250) — compile-verified
//
#include <hip/hip_runtime.h>

// Wiener filter (7x7 box, reflect pad) for [16,3,1024,1024] f32 on MI455X.
//
// Memory-bound: ~30 flops/pixel vs 23.3 TB/s HBM. Two-phase plan exploiting
// the 192 MB L2 (input fits entirely):
//   pass1: per-16x16-tile local variance -> partial sums in d_ws (no big
//          intermediates written to HBM)
//   reduce: per-(b,c) noise_power (fixed-order -> deterministic)
//   pass2: recompute box filter (reads hit L2) and write output.
// The separable 7-tap box filter is computed per-wave with
// V_WMMA_F32_16X16X4_F32 as a banded-matrix multiply:
//   T(24x16) = Xp(24x24) * Wh(24x16);  Out(16x16) = Wv(16x24) * T(24x16)
// with Wh/Wv 0/1 band matrices synthesized ONCE per tile from lane indices,
// then PINNED in VGPRs via empty inline-asm so the compiler cannot
// rematerialize the cmp/cndmask chains in the WMMA inner loop.

typedef __attribute__((ext_vector_type(2))) float v2f;
typedef __attribute__((ext_vector_type(8))) float v8f;

#define HH 1024
#define WW 1024
#define TILES_PER_PLANE 4096   // (1024/16)^2
#define XS 25                  // Xp row stride (24 cols used, +1 pad)
#define TS 17                  // Tb row stride (16 cols used, +1 pad)
#define XP_ELEMS (24 * XS)     // 600 floats
#define TB_ELEMS (24 * TS)     // 408 floats
#define WAVE_LDS (XP_ELEMS + TB_ELEMS)  // 1008 floats per wave

__device__ __forceinline__ v8f wmma4(v2f a, v2f b, v8f c) {
  // D(16x16,f32) = A(16x4,f32) x B(4x16,f32) + C
  return __builtin_amdgcn_wmma_f32_16x16x4_f32(
      /*neg_a=*/false, a, /*neg_b=*/false, b,
      /*c_mod=*/(short)0, c, /*reuse_a=*/false, /*reuse_b=*/false);
}

__device__ __forceinline__ void lds_fence() {
  // cross-lane LDS RAW within the wave: wait on the CDNA5 DS counter
  asm volatile("s_wait_dscnt 0" ::: "memory");
}

// band weight: 1 if n <= k <= n+6 (7-wide summing band)
__device__ __forceinline__ float bandw(int k, int n) {
  return (k >= n && k <= n + 6) ? 1.0f : 0.0f;
}

// Raw 7x7 box SUM of the 16x16 tile whose padded data sits in Xp[24x24].
// wfrag[i] holds the band-matrix fragment for K-step i (pinned in VGPRs).
// Returns D in WMMA C/D layout: lane (n,h), vgpr r -> pixel row r+8h, col n.
__device__ __forceinline__ v8f box49(const float* Xp, float* Tb, int lane,
                                     const v2f wfrag[6]) {
  const int n = lane & 15;
  const int h = lane >> 4;
  // Stage 1 (horizontal): T = Xp * Wh, two M-tiles (rows 0..15 and 8..23)
#pragma unroll
  for (int mt = 0; mt < 2; ++mt) {
    const int M0 = mt * 8;
    v8f acc = {};
#pragma unroll
    for (int i = 0; i < 6; ++i) {
      const int K = 4 * i + 2 * h;
      v2f a;
      a.x = Xp[(M0 + n) * XS + K];       // A frag: M = M0+n, K = K0+2h+j
      a.y = Xp[(M0 + n) * XS + K + 1];
      acc = wmma4(a, wfrag[i], acc);      // B frag: Wh[K][n] (pinned)
    }
#pragma unroll
    for (int r = 0; r < 8; ++r)           // D -> Tb rows M0+r+8h, col n
      Tb[(M0 + r + 8 * h) * TS + n] = acc[r];
  }
  lds_fence();
  // Stage 2 (vertical): Out = Wv * T  (Wv A-frag == Wh B-frag values)
  v8f acc = {};
#pragma unroll
  for (int i = 0; i < 6; ++i) {
    const int K = 4 * i + 2 * h;
    v2f b;
    b.x = Tb[K * TS + n];                 // B frag: T[K][n]
    b.y = Tb[(K + 1) * TS + n];
    acc = wmma4(wfrag[i], b, acc);
  }
  return acc;
}

// Computes local mean/var for one 16x16 tile; optionally the center x values.
__device__ __forceinline__ void wiener_tile(
    const float* __restrict__ xplane, float* Xp, float* Tb,
    int lane, int row0, int col0, bool want_x,
    float mean[8], float var[8], float xv[8]) {
  // Load 22x22 reflect-padded halo tile into 24x24 (zero-filled pad rows/cols;
  // the band weights never touch K=22,23 so zeros are inert).
  for (int idx = lane; idx < 576; idx += 32) {
    const int r = idx / 24;
    const int c = idx - r * 24;
    float v = 0.0f;
    if (r < 22 && c < 22) {
      int gr = row0 - 3 + r;
      gr = gr < 0 ? -gr : (gr > HH - 1 ? 2 * (HH - 1) - gr : gr);
      int gc = col0 - 3 + c;
      gc = gc < 0 ? -gc : (gc > WW - 1 ? 2 * (WW - 1) - gc : gc);
      v = xplane[gr * WW + gc];
    }
    Xp[r * XS + c] = v;
  }
  const int n = lane & 15;
  const int h = lane >> 4;
  // Build band-matrix fragments once, then pin them in VGPRs: the empty asm
  // makes the values opaque so the compiler cannot rematerialize the
  // cmp/cndmask chains next to every WMMA.
  v2f wfrag[6];
#pragma unroll
  for (int i = 0; i < 6; ++i) {
    const int K = 4 * i + 2 * h;
    wfrag[i].x = bandw(K, n);
    wfrag[i].y = bandw(K + 1, n);
    asm volatile("" : "+v"(wfrag[i]));
  }
  lds_fence();
  if (want_x) {
#pragma unroll
    for (int r = 0; r < 8; ++r)
      xv[r] = Xp[(3 + r + 8 * h) * XS + 3 + n];
  }
  v8f Sx = box49(Xp, Tb, lane, wfrag);
  lds_fence();
  // Square the tile in place, rerun the same WMMA pipeline for sum(x^2).
  for (int idx = lane; idx < 576; idx += 32) {
    const int r = idx / 24;
    const int c = idx - r * 24;
    const float t = Xp[r * XS + c];
    Xp[r * XS + c] = t * t;
  }
  lds_fence();
  v8f Sxx = box49(Xp, Tb, lane, wfrag);
  const float inv49 = 1.0f / 49.0f;
#pragma unroll
  for (int r = 0; r < 8; ++r) {
    const float m = Sx[r] * inv49;
    mean[r] = m;
    var[r] = Sxx[r] * inv49 - m * m;
  }
}

// Pass 1: per-tile sum of local variance -> fixed slot (deterministic).
__global__ __launch_bounds__(256) void wiener_pass1(
    const float* __restrict__ x, float* __restrict__ partials) {
  __shared__ float smem[8 * WAVE_LDS];
  const int wave = threadIdx.x >> 5;
  const int lane = threadIdx.x & 31;
  float* Xp = smem + wave * WAVE_LDS;
  float* Tb = Xp + XP_ELEMS;
  const int tile = blockIdx.x * 8 + wave;
  const int plane = tile >> 12;
  const int t = tile & (TILES_PER_PLANE - 1);
  const int row0 = (t >> 6) << 4;
  const int col0 = (t & 63) << 4;
  float mean[8], var[8], xv[8];
  wiener_tile(x + (size_t)plane * (HH * WW), Xp, Tb, lane, row0, col0, false,
              mean, var, xv);
  float s = 0.0f;
#pragma unroll
  for (int r = 0; r < 8; ++r) s += var[r];
#pragma unroll
  for (int off = 16; off > 0; off >>= 1) s += __shfl_down(s, off, 32);
  if (lane == 0) partials[tile] = s;
}

// Per-plane fixed-order reduction -> noise_power = mean(local_var).
__global__ __launch_bounds__(256) void wiener_reduce_np(
    const float* __restrict__ partials, float* __restrict__ noise_power) {
  __shared__ float s[256];
  const float* p = partials + (size_t)blockIdx.x * TILES_PER_PLANE;
  float acc = 0.0f;
  for (int i = threadIdx.x; i < TILES_PER_PLANE; i += 256) acc += p[i];
  s[threadIdx.x] = acc;
  __syncthreads();
  for (int off = 128; off > 0; off >>= 1) {
    if (threadIdx.x < off) s[threadIdx.x] += s[threadIdx.x + off];
    __syncthreads();
  }
  if (threadIdx.x == 0)
    noise_power[blockIdx.x] = s[0] * (1.0f / ((float)HH * (float)WW));
}

// Pass 2: recompute filter (x re-reads mostly hit the 192MB L2), emit output.
__global__ __launch_bounds__(256) void wiener_pass2(
    const float* __restrict__ x, const float* __restrict__ noise_power,
    float* __restrict__ out) {
  __shared__ float smem[8 * WAVE_LDS];
  const int wave = threadIdx.x >> 5;
  const int lane = threadIdx.x & 31;
  float* Xp = smem + wave * WAVE_LDS;
  float* Tb = Xp + XP_ELEMS;
  const int tile = blockIdx.x * 8 + wave;
  const int plane = tile >> 12;
  const int t = tile & (TILES_PER_PLANE - 1);
  const int row0 = (t >> 6) << 4;
  const int col0 = (t & 63) << 4;
  float mean[8], var[8], xv[8];
  wiener_tile(x + (size_t)plane * (HH * WW), Xp, Tb, lane, row0, col0, true,
              mean, var, xv);
  const float np = noise_power[plane];
  const int n = lane & 15;
  const int h = lane >> 4;
  float* oplane = out + (size_t)plane * (HH * WW);
#pragma unroll
  for (int r = 0; r < 8; ++r) {
    const float m = mean[r];
    const float v = var[r];
    const float o = (v < np) ? m : (xv[r] - m) * (1.0f - np / v) + m;
    oplane[(size_t)(row0 + r + 8 * h) * WW + (col0 + n)] = o;
  }
}

extern "C" void kernel_launch(void* const* d_in, const int* in_sizes, int n_in,
                              void* d_out, int out_size, void* d_ws,
                              size_t ws_size, hipStream_t stream) {
  (void)n_in; (void)out_size; (void)ws_size;
  const float* x = (const float*)d_in[0];
  // d_in[1] is ksize == 7 for this config (kernel is specialized for 7).
  float* out = (float*)d_out;
  const int planes = in_sizes[0] / (HH * WW);         // 16*3 = 48
  float* partials = (float*)d_ws;                     // planes*4096 floats
  float* np = partials + (size_t)planes * TILES_PER_PLANE;  // planes floats
  const int tiles = planes * TILES_PER_PLANE;
  wiener_pass1<<<tiles / 8, 256, 0, stream>>>(x, partials);
  wiener_reduce_np<<<planes, 256, 0, stream>>>(partials, np);
  wiener_pass2<<<tiles / 8, 256, 0, stream>>>(x, np, out);
}